// Sampler_25065429139769
// MI455X (gfx1250) — compile-verified
//
#include <hip/hip_runtime.h>
#include <cstdint>
#include <cstddef>

typedef __attribute__((ext_vector_type(4))) float v4f;

#define NB    256        // batch rows
#define NV    128000     // vocab
#define TPB   256        // threads per block (8 wave32 waves)
#define TILE  1024       // floats per staged tile (TPB * 4)
#define NITER (NV / TILE) // 125

// CDNA5 async global->LDS staging (ASYNCcnt path). Each lane stages its own
// 16B slot, so a per-wave s_wait_asynccnt suffices (no cross-lane LDS deps).
__device__ __forceinline__ void async_ld16(uint32_t lds, const float* g) {
    asm volatile("global_load_async_to_lds_b128 %0, %1, off"
                 :: "v"(lds), "v"((uint64_t)(uintptr_t)g) : "memory");
}
// Non-temporal variant for the use-once noise stream: keep L2 free for logits
// so the pass-2 re-read of logits hits the 192MB L2 instead of HBM.
__device__ __forceinline__ void async_ld16_nt(uint32_t lds, const float* g) {
    asm volatile("global_load_async_to_lds_b128 %0, %1, off th:TH_LOAD_NT"
                 :: "v"(lds), "v"((uint64_t)(uintptr_t)g) : "memory");
}

__global__ __launch_bounds__(TPB) void sampler_kernel(
    const float* __restrict__ logits,
    const float* __restrict__ temps,
    const float* __restrict__ noise,
    float* __restrict__ out)
{
    __shared__ float sLog[2][TILE];
    __shared__ float sNoi[2][TILE];
    __shared__ float rMaxL[TPB];
    __shared__ int   rIdxL[TPB];
    __shared__ float rMaxS[TPB];
    __shared__ int   rIdxS[TPB];
    __shared__ float rM[TPB];
    __shared__ float rS[TPB];
    __shared__ float bc[2];

    const int row = blockIdx.x;
    const int tid = threadIdx.x;

    const float t    = temps[row];
    const float invT = (t == 0.0f) ? 1.0f : (1.0f / t);

    const float* lrow = logits + (size_t)row * NV;
    const float* nrow = noise  + (size_t)row * NV;

    // LDS byte offsets of this thread's private 16B slot in each buffer.
    // Generic-pointer low 32 bits == wave-relative LDS byte address (ISA aperture rule).
    const uint32_t ldsL0 = (uint32_t)(uintptr_t)&sLog[0][tid * 4];
    const uint32_t ldsL1 = (uint32_t)(uintptr_t)&sLog[1][tid * 4];
    const uint32_t ldsN0 = (uint32_t)(uintptr_t)&sNoi[0][tid * 4];
    const uint32_t ldsN1 = (uint32_t)(uintptr_t)&sNoi[1][tid * 4];

    // ---- Pass 1: single streaming read; fused greedy / gumbel / online-softmax ----
    async_ld16   (ldsL0, lrow + tid * 4);
    async_ld16_nt(ldsN0, nrow + tid * 4);

    float maxL = -__builtin_inff(); int idxL = 0;   // greedy argmax
    float maxS = -__builtin_inff(); int idxS = 0;   // gumbel-max argmax
    float m    = -__builtin_inff(); float ssum = 0.0f; // online softmax (max, sum)

    for (int it = 0; it < NITER; ++it) {
        const int cur = it & 1;
        if (it + 1 < NITER) {
            const int base = (it + 1) * TILE + tid * 4;
            // Close the DScnt/ASYNCcnt cross-counter ordering hole before
            // overwriting the other buffer (different counters are unordered).
            asm volatile("s_wait_dscnt 0x0" ::: "memory");
            if (cur) { async_ld16(ldsL0, lrow + base); async_ld16_nt(ldsN0, nrow + base); }
            else     { async_ld16(ldsL1, lrow + base); async_ld16_nt(ldsN1, nrow + base); }
            // Two newest async ops may stay in flight; wait for current tile only.
            asm volatile("s_wait_asynccnt 0x2" ::: "memory");
        } else {
            asm volatile("s_wait_asynccnt 0x0" ::: "memory");
        }

        const v4f l4 = *(const v4f*)&sLog[cur][tid * 4];
        const v4f u4 = *(const v4f*)&sNoi[cur][tid * 4];
        const int jb = it * TILE + tid * 4;
#pragma unroll
        for (int k = 0; k < 4; ++k) {
            const float l = l4[k];
            const float u = u4[k];
            const int   j = jb + k;
            if (l > maxL) { maxL = l; idxL = j; }          // first-occurrence max
            const float sc = l * invT;
            const float g  = -__logf(-__logf(u));          // Gumbel(0,1)
            const float sv = sc + g;
            if (sv > maxS) { maxS = sv; idxS = j; }
            if (sc > m) { ssum = ssum * __expf(m - sc) + 1.0f; m = sc; }
            else        { ssum += __expf(sc - m); }
        }
    }

    // ---- Block tree reduction (wave32-safe, no warp-size assumptions) ----
    rMaxL[tid] = maxL; rIdxL[tid] = idxL;
    rMaxS[tid] = maxS; rIdxS[tid] = idxS;
    rM[tid] = m;       rS[tid] = ssum;
    __syncthreads();
    for (int s = TPB / 2; s > 0; s >>= 1) {
        if (tid < s) {
            float a = rMaxL[tid], b2 = rMaxL[tid + s];
            int  ia = rIdxL[tid], ib = rIdxL[tid + s];
            if (b2 > a || (b2 == a && ib < ia)) { rMaxL[tid] = b2; rIdxL[tid] = ib; }

            a  = rMaxS[tid]; b2 = rMaxS[tid + s];
            ia = rIdxS[tid]; ib = rIdxS[tid + s];
            if (b2 > a || (b2 == a && ib < ia)) { rMaxS[tid] = b2; rIdxS[tid] = ib; }

            const float m1 = rM[tid], m2 = rM[tid + s];
            const float s1 = rS[tid], s2 = rS[tid + s];
            const float M  = fmaxf(m1, m2);
            rM[tid] = M;
            rS[tid] = s1 * __expf(m1 - M) + s2 * __expf(m2 - M);
        }
        __syncthreads();
    }
    if (tid == 0) {
        const int token = (t == 0.0f) ? rIdxL[0] : rIdxS[0];
        out[row] = (float)token;          // token exactly representable (V < 2^24)
        bc[0] = rM[0];                    // global max of scaled logits
        bc[1] = 1.0f / rS[0];             // reciprocal softmax denominator
    }
    __syncthreads();
    const float M    = bc[0];
    const float rden = bc[1];

    // ---- Pass 2: re-read logits (L2-resident), write probs non-temporally ----
    float* prow = out + NB + (size_t)row * NV;
    for (int it = 0; it < NITER; ++it) {
        const int q = it * TILE + tid * 4;
        const v4f l4 = *(const v4f*)(lrow + q);
        v4f p;
#pragma unroll
        for (int k = 0; k < 4; ++k)
            p[k] = __expf(l4[k] * invT - M) * rden;
        __builtin_nontemporal_store(p, (v4f*)(prow + q));
    }
}

extern "C" void kernel_launch(void* const* d_in, const int* in_sizes, int n_in,
                              void* d_out, int out_size, void* d_ws, size_t ws_size,
                              hipStream_t stream) {
    (void)in_sizes; (void)n_in; (void)out_size; (void)d_ws; (void)ws_size;
    const float* logits = (const float*)d_in[0];
    const float* temps  = (const float*)d_in[1];
    const float* noise  = (const float*)d_in[2];
    float* out = (float*)d_out;
    hipLaunchKernelGGL(sampler_kernel, dim3(NB), dim3(TPB), 0, stream,
                       logits, temps, noise, out);
}